// NCEAverageMultiview_23081154248915
// MI455X (gfx1250) — compile-verified
//
#include <hip/hip_runtime.h>
#include <math.h>

#define B_SZ 256
#define KP1  4097
#define N_SZ 1000000
#define D_SZ 128
#define INV_T (1.0f / 0.07f)

typedef float v2f __attribute__((ext_vector_type(2)));
typedef float v8f __attribute__((ext_vector_type(8)));
typedef float f32x4 __attribute__((ext_vector_type(4)));
typedef int   i32x4 __attribute__((vector_size(16)));

typedef __attribute__((address_space(1))) i32x4* gptr_i32x4;  // global
typedef __attribute__((address_space(3))) i32x4* lptr_i32x4;  // LDS

#define HAVE_ASYNC_LDS __has_builtin(__builtin_amdgcn_global_load_async_to_lds_b128)

// ---------------------------------------------------------------------------
// Dot kernel: out[b,k] = exp(dot(mem[row(b,k)], v[b]) / T)  via fp32 WMMA.
// grid = (65, 256, 2); block = 128 (4 waves). Wave w handles 16 k-values.
// view 0: out_v1 = f(memory_v2, v1); view 1: out_v2 = f(memory_v1, v2).
// Gathered rows are staged into LDS with GLOBAL_LOAD_ASYNC_TO_LDS_B128
// (CDNA5 async copy engine, ASYNCcnt-tracked) when the builtin is available.
// ---------------------------------------------------------------------------
__global__ __launch_bounds__(128) void nce_dot(
    const float* __restrict__ v1, const float* __restrict__ v2,
    const float* __restrict__ mem1, const float* __restrict__ mem2,
    const int* __restrict__ y, const int* __restrict__ idx,
    float* __restrict__ out_v1, float* __restrict__ out_v2)
{
    const int view = blockIdx.z;
    const int b    = blockIdx.y;
    const int wave = threadIdx.x >> 5;
    const int lane = threadIdx.x & 31;
    const int kbase = blockIdx.x * 64 + wave * 16;

    const float* __restrict__ v   = (view == 0) ? v1 : v2;
    const float* __restrict__ mem = (view == 0) ? mem2 : mem1;
    float* __restrict__ out       = (view == 0) ? out_v1 : out_v2;

    __shared__ float s_v[D_SZ];
    __shared__ __align__(16) float s_rows[4][16][132]; // 132-float stride: conflict-free b64 reads

    // stage v[b] (128 floats, one per thread)
    s_v[threadIdx.x] = v[b * D_SZ + threadIdx.x];

    // stage 16 gathered rows per wave: 32 lanes x 16B = one 512B row per issue
    #pragma unroll
    for (int r = 0; r < 16; ++r) {
        int k   = kbase + r;
        int row = 0;
        if (k < KP1) row = (k == 0) ? y[b] : idx[b * KP1 + k];  // idx[:,0] := y
        const float* src = mem + (size_t)row * D_SZ + lane * 4;
#if HAVE_ASYNC_LDS
        __builtin_amdgcn_global_load_async_to_lds_b128(
            (gptr_i32x4)(void*)src,
            (lptr_i32x4)(void*)&s_rows[wave][r][lane * 4],
            0, 0);
#else
        *(f32x4*)&s_rows[wave][r][lane * 4] = *(const f32x4*)src;
#endif
    }
#if HAVE_ASYNC_LDS
#if __has_builtin(__builtin_amdgcn_s_wait_asynccnt)
    __builtin_amdgcn_s_wait_asynccnt(0);
#else
    asm volatile("s_wait_asynccnt 0x0" ::: "memory");
#endif
#endif
    __syncthreads();

    // A fragment: lane m<16 -> row m, K={d0,d0+1}; lanes 16-31 -> row m-16, K={d0+2,d0+3}
    // B fragment: v[b] broadcast into all 16 columns (same K striping assumed).
    const int m  = lane & 15;
    const int hi = (lane >> 4) << 1;       // 0 or 2
    v8f c = {0.f, 0.f, 0.f, 0.f, 0.f, 0.f, 0.f, 0.f};
    #pragma unroll
    for (int d0 = 0; d0 < D_SZ; d0 += 4) {
        v2f a  = *(const v2f*)&s_rows[wave][m][d0 + hi];
        v2f bm = *(const v2f*)&s_v[d0 + hi];
        c = __builtin_amdgcn_wmma_f32_16x16x4_f32(false, a, false, bm,
                                                  (short)0, c, false, false);
    }

    // All 16 output columns are identical: lane 0 holds M=0..7, lane 16 holds M=8..15.
    if ((lane & 15) == 0) {
        const int mb = kbase + ((lane >> 4) << 3);
        #pragma unroll
        for (int r = 0; r < 8; ++r) {
            int k = mb + r;
            if (k < KP1) out[(size_t)b * KP1 + k] = expf(c[r] * INV_T);
        }
    }
}

// ---------------------------------------------------------------------------
// Deterministic Z reduction: one block per view, fixed-order strided sum.
// ---------------------------------------------------------------------------
__global__ __launch_bounds__(256) void nce_reduce(const float* __restrict__ outs,
                                                  float* __restrict__ zsum)
{
    const float* p = outs + (size_t)blockIdx.x * (B_SZ * KP1);
    __shared__ float red[256];
    float local = 0.f;
    for (int i = threadIdx.x; i < B_SZ * KP1; i += 256) local += p[i];
    red[threadIdx.x] = local;
    __syncthreads();
    for (int s = 128; s > 0; s >>= 1) {
        if (threadIdx.x < s) red[threadIdx.x] += red[threadIdx.x + s];
        __syncthreads();
    }
    if (threadIdx.x == 0) zsum[blockIdx.x] = red[0];
}

// out /= (mean * N)  where mean = zsum / (B*(K+1))
__global__ void nce_norm(float* __restrict__ outs, const float* __restrict__ zsum)
{
    const size_t total = (size_t)2 * B_SZ * KP1;
    for (size_t i = (size_t)blockIdx.x * blockDim.x + threadIdx.x; i < total;
         i += (size_t)gridDim.x * blockDim.x) {
        int view = (i >= (size_t)(B_SZ * KP1)) ? 1 : 0;
        float Z  = zsum[view] * ((float)N_SZ / (float)(B_SZ * KP1));
        outs[i] = outs[i] / Z;
    }
}

// bulk bank copy: regular loads, non-temporal stores (don't pollute L2 — the
// gather kernel that follows wants L2 for the random row working set).
__global__ void nce_copy(const f32x4* __restrict__ src, f32x4* __restrict__ dst,
                         size_t n4)
{
    for (size_t i = (size_t)blockIdx.x * blockDim.x + threadIdx.x; i < n4;
         i += (size_t)gridDim.x * blockDim.x) {
        f32x4 val = src[i];
        __builtin_nontemporal_store(val, &dst[i]);
    }
}

// momentum + L2-normalize update of rows y[b]; one block per (view, b)
__global__ __launch_bounds__(128) void nce_update(
    const float* __restrict__ v1, const float* __restrict__ v2,
    const float* __restrict__ mem1, const float* __restrict__ mem2,
    const int* __restrict__ y,
    float* __restrict__ omem1, float* __restrict__ omem2)
{
    const int view = blockIdx.x >> 8;
    const int b    = blockIdx.x & 255;
    const int t    = threadIdx.x;
    const float* v   = view ? v2 : v1;
    const float* mem = view ? mem2 : mem1;
    float* om        = view ? omem2 : omem1;

    const int row = y[b];
    float p = mem[(size_t)row * D_SZ + t] * 0.5f + v[b * D_SZ + t] * 0.5f;

    __shared__ float red[128];
    red[t] = p * p;
    __syncthreads();
    for (int s = 64; s > 0; s >>= 1) {
        if (t < s) red[t] += red[t + s];
        __syncthreads();
    }
    om[(size_t)row * D_SZ + t] = p / sqrtf(red[0]);
}

extern "C" void kernel_launch(void* const* d_in, const int* in_sizes, int n_in,
                              void* d_out, int out_size, void* d_ws, size_t ws_size,
                              hipStream_t stream)
{
    const float* v1   = (const float*)d_in[0];
    const float* v2   = (const float*)d_in[1];
    const float* mem1 = (const float*)d_in[2];
    const float* mem2 = (const float*)d_in[3];
    const int*   y    = (const int*)d_in[4];
    const int*   idx  = (const int*)d_in[5];

    float* out    = (float*)d_out;
    float* out_v1 = out;
    float* out_v2 = out + (size_t)B_SZ * KP1;
    float* omem1  = out + (size_t)2 * B_SZ * KP1;
    float* omem2  = omem1 + (size_t)N_SZ * D_SZ;
    float* zsum   = (float*)d_ws;           // 2 floats, rewritten each call

    const size_t n4 = (size_t)N_SZ * D_SZ / 4;
    nce_copy<<<8192, 256, 0, stream>>>((const f32x4*)mem1, (f32x4*)omem1, n4);
    nce_copy<<<8192, 256, 0, stream>>>((const f32x4*)mem2, (f32x4*)omem2, n4);

    dim3 grid((KP1 + 63) / 64, B_SZ, 2);    // 65 x 256 x 2
    nce_dot<<<grid, 128, 0, stream>>>(v1, v2, mem1, mem2, y, idx, out_v1, out_v2);

    nce_reduce<<<2, 256, 0, stream>>>(out_v1, zsum);
    nce_norm<<<512, 256, 0, stream>>>(out_v1, zsum);
    nce_update<<<512, 128, 0, stream>>>(v1, v2, mem1, mem2, y, omem1, omem2);
}